// Agent_BC_MB_30829275250944
// MI455X (gfx1250) — compile-verified
//
#include <hip/hip_runtime.h>

typedef float v2f __attribute__((ext_vector_type(2)));
typedef float v8f __attribute__((ext_vector_type(8)));

#define NMODES 16
#define DIN    10
#define DROOT  32
#define DHID   16
#define WAVES_PER_BLOCK 8
#define TILE   16
#define NBLOCKS 512

__launch_bounds__(256, 1)
__global__ void moe_actor_wmma(const float* __restrict__ obs,
                               const int*   __restrict__ zmode,
                               const float* __restrict__ W0,
                               const float* __restrict__ b0,
                               const float* __restrict__ Wx1, const float* __restrict__ bx1,
                               const float* __restrict__ Wx2, const float* __restrict__ bx2,
                               const float* __restrict__ Wy1, const float* __restrict__ by1,
                               const float* __restrict__ Wy2, const float* __restrict__ by2,
                               float* __restrict__ out, int ntok)
{
    // Expert weights packed as float2 pairs over the hidden dim, with the
    // (head*16 + expert) column as the fastest-varying *pair* index:
    //   sW1[((d*8 + h2)*32 + col)*2 + (h&1)]
    // For fixed (d,h2) the 32 lanes' dword addresses differ by col*2 -> they
    // cover exactly the 64 LDS banks (conflict-free; same-expert lanes broadcast).
    __shared__ float sW1[DROOT * (DHID/2) * 32 * 2];    // 64 KB
    __shared__ float sB1[(DHID/2) * 32 * 2];
    __shared__ float sW2[(DHID/2) * 32 * 2];            // layer-2 col 0 (loc) only
    __shared__ float sB2[32];
    __shared__ float sVec[WAVES_PER_BLOCK][TILE][36];   // per-wave trunk tile, padded rows

    const int tid = threadIdx.x;

    // ---- stage all expert weights into LDS once per block ----
    for (int i = tid; i < NMODES * DROOT * DHID; i += 256) {   // 8192 elems per head
        int e  = i >> 9;            // / (32*16)
        int d  = (i >> 4) & 31;
        int h  = i & 15;
        int o  = ((d * (DHID/2) + (h >> 1)) * 32 + e) * 2 + (h & 1);
        sW1[o]      = Wx1[i];       // head 0 -> col e
        sW1[o + 32] = Wy1[i];       // head 1 -> col 16+e  (pair stride = 2 floats)
    }
    for (int i = tid; i < NMODES * DHID; i += 256) {           // 256 elems per head
        int e = i >> 4, h = i & 15;
        int o = ((h >> 1) * 32 + e) * 2 + (h & 1);
        sB1[o]      = bx1[i];
        sB1[o + 32] = by1[i];
        sW2[o]      = Wx2[2 * i];   // only column 0 (loc) is ever used
        sW2[o + 32] = Wy2[2 * i];
    }
    if (tid < NMODES) {
        sB2[tid]      = bx2[2 * tid];
        sB2[16 + tid] = by2[2 * tid];
    }
    __syncthreads();

    const int lane = tid & 31;
    const int lhi  = lane >> 4;    // 0/1: selects K-pair for WMMA frags, head for experts
    const int ll   = lane & 15;
    const int w    = tid >> 5;

    // ---- hoist trunk B-fragments (W0 constant across tiles) ----
    // Layout mirrors documented 16x4 f32 A layout: VGPR0 = {K=k0 | K=k0+2},
    // VGPR1 = {K=k0+1 | K=k0+3} across lane halves; N = lane%16 (+16 for 2nd tile).
    // K >= 10 entries are ZERO here, which lets the A side load arbitrary finite
    // data for those K without affecting the product.
    v2f Bf[3][2];
    #pragma unroll
    for (int ks = 0; ks < 3; ++ks) {
        int k0 = ks * 4 + lhi * 2;
        #pragma unroll
        for (int nt = 0; nt < 2; ++nt) {
            int n = nt * 16 + ll;
            Bf[ks][nt].x = (k0     < DIN) ? W0[k0 * DROOT + n]       : 0.f;
            Bf[ks][nt].y = (k0 + 1 < DIN) ? W0[(k0 + 1) * DROOT + n] : 0.f;
        }
    }
    const float bb0 = b0[ll];
    const float bb1 = b0[16 + ll];

    const int ntiles  = (ntok + TILE - 1) / TILE;
    const int wstride = gridDim.x * WAVES_PER_BLOCK;

    const float2* w1c = (const float2*)sW1;
    const float2* b1c = (const float2*)sB1;
    const float2* w2c = (const float2*)sW2;

    for (int tile = blockIdx.x * WAVES_PER_BLOCK + w; tile < ntiles; tile += wstride) {
        const int t0 = tile * TILE;
        if (tile + wstride < ntiles)   // prefetch next tile's obs rows (global_prefetch_b8)
            __builtin_prefetch(&obs[(size_t)(t0 + wstride * TILE + ll) * DIN], 0, 1);

        int row = t0 + ll; if (row >= ntok) row = ntok - 1;
        const float* arow = obs + (size_t)row * DIN;

        // ---- trunk: vec[16x32] = relu(obs[16x10] @ W0[10x32] + b0) via WMMA f32 ----
        v8f c0 = {bb0, bb0, bb0, bb0, bb0, bb0, bb0, bb0};
        v8f c1 = {bb1, bb1, bb1, bb1, bb1, bb1, bb1, bb1};
        #pragma unroll
        for (int ks = 0; ks < 3; ++ks) {
            int k0 = ks * 4 + lhi * 2;
            // Out-of-range K (only ks==2, upper lane half): re-load the K=8..9
            // pair instead; matching B fragment lanes are zero, so the product
            // contributes nothing. Keeps the load unconditional (no EXEC masking).
            int koff = (k0 + 1 < DIN) ? k0 : (DIN - 2);
            v2f a = *(const v2f*)(arow + koff);      // 8B-aligned global_load_b64
            c0 = __builtin_amdgcn_wmma_f32_16x16x4_f32(false, a, false, Bf[ks][0],
                                                       (short)0, c0, false, false);
            c1 = __builtin_amdgcn_wmma_f32_16x16x4_f32(false, a, false, Bf[ks][1],
                                                       (short)0, c1, false, false);
        }

        // ---- ReLU + transpose through per-wave LDS tile ----
        // C layout: VGPR r holds row (r + 8*lane_half), col = lane%16 (+16 for c1).
        #pragma unroll
        for (int r = 0; r < 8; ++r) {
            sVec[w][r + 8 * lhi][ll]      = fmaxf(c0[r], 0.f);
            sVec[w][r + 8 * lhi][16 + ll] = fmaxf(c1[r], 0.f);
        }
        asm volatile("s_wait_dscnt 0" ::: "memory");   // wave-internal LDS RAW fence

        // ---- expert MLP: lane -> (token = ll, head = lhi) ----
        const int gtok = t0 + ll;
        const int zt   = zmode[(gtok < ntok) ? gtok : (ntok - 1)] & (NMODES - 1);
        const int col  = lhi * 16 + zt;

        const float4* vrow = (const float4*)&sVec[w][ll][0];   // 144B row offset: 16B aligned
        float v[DROOT];
        #pragma unroll
        for (int q = 0; q < 8; ++q) {
            float4 t4 = vrow[q];
            v[4*q+0] = t4.x; v[4*q+1] = t4.y; v[4*q+2] = t4.z; v[4*q+3] = t4.w;
        }

        float acc[DHID];
        #pragma unroll
        for (int h2 = 0; h2 < DHID/2; ++h2) {
            float2 bb = b1c[h2 * 32 + col];
            acc[2*h2]     = bb.x;
            acc[2*h2 + 1] = bb.y;
        }
        #pragma unroll 4
        for (int d = 0; d < DROOT; ++d) {
            float vd = v[d];
            #pragma unroll
            for (int h2 = 0; h2 < DHID/2; ++h2) {
                float2 ww = w1c[(d * (DHID/2) + h2) * 32 + col];   // ds_load_b64
                acc[2*h2]     = fmaf(vd, ww.x, acc[2*h2]);
                acc[2*h2 + 1] = fmaf(vd, ww.y, acc[2*h2 + 1]);
            }
        }
        float res = sB2[col];
        #pragma unroll
        for (int h2 = 0; h2 < DHID/2; ++h2) {
            float2 ww = w2c[h2 * 32 + col];
            res = fmaf(fmaxf(acc[2*h2],     0.f), ww.x, res);
            res = fmaf(fmaxf(acc[2*h2 + 1], 0.f), ww.y, res);
        }

        if (gtok < ntok) out[(size_t)gtok * 2 + lhi] = res;

        asm volatile("s_wait_dscnt 0" ::: "memory");   // WAR fence before next tile reuse
    }
}

extern "C" void kernel_launch(void* const* d_in, const int* in_sizes, int n_in,
                              void* d_out, int out_size, void* d_ws, size_t ws_size,
                              hipStream_t stream) {
    const float* obs = (const float*)d_in[0];
    const int*   z   = (const int*)  d_in[1];
    const float* W0  = (const float*)d_in[2];
    const float* b0  = (const float*)d_in[3];
    const float* Wx1 = (const float*)d_in[4];
    const float* bx1 = (const float*)d_in[5];
    const float* Wx2 = (const float*)d_in[6];
    const float* bx2 = (const float*)d_in[7];
    const float* Wy1 = (const float*)d_in[8];
    const float* by1 = (const float*)d_in[9];
    const float* Wy2 = (const float*)d_in[10];
    const float* by2 = (const float*)d_in[11];
    float* out = (float*)d_out;

    const int ntok   = in_sizes[0] / DIN;                      // 262144
    const int ntiles = (ntok + TILE - 1) / TILE;               // 16384
    int blocks = NBLOCKS;                                      // 4096 waves, 4 tiles/wave
    int maxb = (ntiles + WAVES_PER_BLOCK - 1) / WAVES_PER_BLOCK;
    if (blocks > maxb) blocks = maxb;
    if (blocks < 1) blocks = 1;

    moe_actor_wmma<<<blocks, 256, 0, stream>>>(obs, z, W0, b0,
                                               Wx1, bx1, Wx2, bx2,
                                               Wy1, by1, Wy2, by2,
                                               out, ntok);
}